// ForwardDecoder_16862041604722
// MI455X (gfx1250) — compile-verified
//
#include <hip/hip_runtime.h>
#include <stdint.h>

// ---------------------------------------------------------------------------
// HMM forward DP over anti-diagonals (softmax-operator / logsumexp semiring).
// Latency-bound wavefront: 4095 sequential diagonals, <=2048 cells each.
// Strategy: one persistent workgroup, LDS-resident diagonal state, wave32
// VALU with native v_exp_f32/v_log_f32, and CDNA5 async global->LDS staging
// of theta one diagonal ahead (double-buffered ring) to hide the strided
// (~24KB apart per lane) first-touch latency. WMMA is not applicable: the
// inner op is a 3x3 logsumexp-semiring contraction, not a (+,x) matmul.
// ---------------------------------------------------------------------------

#define N_DIM   2048
#define M_DIM   2048
#define NEG_INF (-1.0e8f)
#define NTHREADS 1024
#define CPT      2                    // cells per thread = N_DIM / NTHREADS
#define VSLOT    4                    // state vector padded 3 -> 4 floats (16B)
#define VBUF_FLOATS  ((N_DIM + 1) * VSLOT)
#define STAGE_FLOATS (N_DIM * VSLOT)
#define LDS_FLOATS   (3 * VBUF_FLOATS + 2 * STAGE_FLOATS)

#if defined(__has_builtin)
# if __has_builtin(__builtin_amdgcn_global_load_async_to_lds_b32)
#  define HAVE_ASYNC_B32 1
# endif
# if __has_builtin(__builtin_amdgcn_s_wait_asynccnt)
#  define HAVE_WAIT_ASYNC 1
# endif
#endif

typedef __attribute__((address_space(1))) int gint_t;   // global (AS1)
typedef __attribute__((address_space(3))) int lint_t;   // LDS    (AS3)

__device__ __forceinline__ void async_copy_b32(const float* gsrc, float* ldst) {
#ifdef HAVE_ASYNC_B32
  __builtin_amdgcn_global_load_async_to_lds_b32(
      (gint_t*)gsrc, (lint_t*)ldst, /*offset=*/0, /*cpol=*/0);
#else
  // VGLOBAL async load, GV mode: vdst = LDS byte address, vaddr = 64-bit addr.
  asm volatile("global_load_async_to_lds_b32 %0, %1, off"
               :: "v"((uint32_t)(uintptr_t)ldst), "v"(gsrc)
               : "memory");
#endif
}

__device__ __forceinline__ void wait_async_le6() {
#ifdef HAVE_WAIT_ASYNC
  __builtin_amdgcn_s_wait_asynccnt(6);
#else
  asm volatile("s_wait_asynccnt 6" ::: "memory");
#endif
}

// Stage theta[(i-1), (d-i)-1, 0..2] for every cell i owned by this thread on
// diagonal d into the LDS ring slot (d & 1). Indices are clipped exactly like
// the reference (invalid cells load in-bounds garbage, later masked NEG_INF),
// so every wave issues exactly 6 async B32 copies -> ASYNCcnt bookkeeping is
// uniform and a "wait <= 6" releases the previous diagonal's data.
__device__ __forceinline__ void stage_theta(const float* __restrict__ theta,
                                            float* __restrict__ stagebase,
                                            int d, int t) {
  float* dst = stagebase + (size_t)(d & 1) * STAGE_FLOATS;
#pragma unroll
  for (int c = 0; c < CPT; ++c) {
    const int i = 1 + t + c * NTHREADS;          // i in [1, N]
    int j = d - i;
    j = (j < 1) ? 1 : ((j > M_DIM) ? M_DIM : j); // clip like jnp.clip
    const float* src = theta + ((size_t)(i - 1) * M_DIM + (size_t)(j - 1)) * 3;
    float* dl = dst + (size_t)(i - 1) * VSLOT;
    async_copy_b32(src + 0, dl + 0);
    async_copy_b32(src + 1, dl + 1);
    async_copy_b32(src + 2, dl + 2);
  }
}

__device__ __forceinline__ float lse3(float x, float y, float z) {
  float m = fmaxf(x, fmaxf(y, z));
  float s = __expf(x - m) + __expf(y - m) + __expf(z - m);
  return m + __logf(s);
}

extern "C" __global__ __launch_bounds__(NTHREADS)
void fwd_dp_kernel(const float* __restrict__ theta,
                   const float* __restrict__ A,
                   float* __restrict__ out) {
  extern __shared__ float lds[];
  float* vbuf  = lds;                          // 3 rotating diagonal buffers
  float* stage = lds + 3 * VBUF_FLOATS;        // 2-deep theta staging ring

  const int t = threadIdx.x;

  // Transition matrix (uniform -> scalar loads).
  const float a00 = A[0], a01 = A[1], a02 = A[2];
  const float a10 = A[3], a11 = A[4], a12 = A[5];
  const float a20 = A[6], a21 = A[7], a22 = A[8];

  // Init: all diagonal buffers to NEG_INF; V[0,0,:] = 0 on the d=0 buffer.
  for (int idx = t; idx < 3 * VBUF_FLOATS; idx += NTHREADS) vbuf[idx] = NEG_INF;
  __syncthreads();
  if (t == 0) { vbuf[0] = 0.0f; vbuf[1] = 0.0f; vbuf[2] = 0.0f; }

  // Prologue: stage theta for the first real diagonal (d = 2).
  stage_theta(theta, stage, 2, t);
  __syncthreads();

  for (int d = 2; d <= N_DIM + M_DIM; ++d) {
    // Prefetch theta for diagonal d+1 while we compute diagonal d.
    stage_theta(theta, stage, d + 1, t);
    // Diagonal d's 6 older async copies (per wave) are complete once ASYNCcnt
    // drops to the 6 just-issued prefetches (async loads retire in order).
    wait_async_le6();

    const float* vp2 = vbuf + (size_t)((d - 2) % 3) * VBUF_FLOATS; // diag d-2
    const float* vp1 = vbuf + (size_t)((d - 1) % 3) * VBUF_FLOATS; // diag d-1
    float*       vnw = vbuf + (size_t)(d % 3) * VBUF_FLOATS;       // diag d
    const float* st  = stage + (size_t)(d & 1) * STAGE_FLOATS;

    // Row-0 border of the new diagonal must be NEG_INF (clears stale V[0,0]).
    if (t == 0) { vnw[0] = NEG_INF; vnw[1] = NEG_INF; vnw[2] = NEG_INF; vnw[3] = NEG_INF; }

#pragma unroll
    for (int c = 0; c < CPT; ++c) {
      const int i = 1 + t + c * NTHREADS;   // 1 <= i <= N always
      const int j = d - i;
      const bool valid = (j >= 1) && (j <= M_DIM);

      const float4 pm = *(const float4*)(vp2 + (size_t)(i - 1) * VSLOT); // V[i-1,j-1]
      const float4 px = *(const float4*)(vp1 + (size_t)(i - 1) * VSLOT); // V[i-1,j]
      const float4 py = *(const float4*)(vp1 + (size_t)i       * VSLOT); // V[i,j-1]
      const float4 th = *(const float4*)(st  + (size_t)(i - 1) * VSLOT); // theta[i-1,j-1,:]

      const float r0 = lse3(pm.x + a00, pm.y + a01, pm.z + a02) + th.x;  // match
      const float r1 = lse3(px.x + a10, px.y + a11, px.z + a12) + th.y;  // insert-x
      const float r2 = lse3(py.x + a20, py.y + a21, py.z + a22) + th.z;  // insert-y

      float4 w;
      w.x = valid ? r0 : NEG_INF;
      w.y = valid ? r1 : NEG_INF;
      w.z = valid ? r2 : NEG_INF;
      w.w = NEG_INF;
      *(float4*)(vnw + (size_t)i * VSLOT) = w;
    }
    __syncthreads();   // publish diagonal d before it becomes vp1
  }

  if (t == 0) {
    const float* vlast = vbuf + (size_t)((N_DIM + M_DIM) % 3) * VBUF_FLOATS;
    const float v0 = vlast[(size_t)N_DIM * VSLOT + 0];
    const float v1 = vlast[(size_t)N_DIM * VSLOT + 1];
    const float v2 = vlast[(size_t)N_DIM * VSLOT + 2];
    out[0] = lse3(v0, v1, v2);
  }
}

extern "C" void kernel_launch(void* const* d_in, const int* in_sizes, int n_in,
                              void* d_out, int out_size, void* d_ws, size_t ws_size,
                              hipStream_t stream) {
  (void)in_sizes; (void)n_in; (void)d_ws; (void)ws_size; (void)out_size;
  const float* theta = (const float*)d_in[0];
  const float* A     = (const float*)d_in[1];
  float*       out   = (float*)d_out;

  const size_t shmem = (size_t)LDS_FLOATS * sizeof(float);   // ~160 KB (< 320 KB/WGP)
  (void)hipFuncSetAttribute(reinterpret_cast<const void*>(&fwd_dp_kernel),
                            hipFuncAttributeMaxDynamicSharedMemorySize, (int)shmem);
  fwd_dp_kernel<<<1, NTHREADS, shmem, stream>>>(theta, A, out);
}